// YOLOHead_86629490360416
// MI455X (gfx1250) — compile-verified
//
#include <hip/hip_runtime.h>
#include <hip/hip_bf16.h>
#include <cstdint>
#include <cstddef>

// ---- problem constants (fixed by the reference) ----
#define NANCH   3
#define NCLS    80
#define ATTRS   85          // 5 + NCLS
#define GH      80
#define GW      80
#define HW      6400        // GH*GW
#define BATCH   32
#define STRIDE_PX 8.0f      // 640/80 for both x and y

// ---- tiling ----
#define TP          128                 // pixels per tile
#define LDS_STRIDE  129                 // TP+1 via TDM pad: conflict-free transposed reads
#define TILES_PP    (HW / TP)           // 50 tiles per (b,anchor) plane
#define BLOCK       256                 // 8 wave32s

typedef uint32_t v4u __attribute__((ext_vector_type(4)));
typedef uint32_t v8u __attribute__((ext_vector_type(8)));

__device__ __forceinline__ float fast_sigmoid(float v) {
    return 1.0f / (1.0f + __expf(-v));
}

__global__ __launch_bounds__(BLOCK)
void yolo_head_kernel(const float* __restrict__ in,
                      const float* __restrict__ anchors,
                      float* __restrict__ out) {
    __shared__ float tile[ATTRS * LDS_STRIDE];   // ~43.9 KB (85 rows, 129-dw stride)

    const uint32_t tid    = threadIdx.x;
    const uint32_t tidx   = blockIdx.x % TILES_PP;
    const uint32_t plane  = blockIdx.x / TILES_PP;
    const uint32_t a      = plane % NANCH;
    const uint32_t b      = plane / NANCH;
    const uint32_t pixel0 = tidx * TP;

    // input tile start for (b, anchor): channel a*85, pixel pixel0
    const float* in_plane =
        in + ((size_t)(b * (NANCH * ATTRS) + a * ATTRS)) * HW + pixel0;

    // ---- Phase 1: one TDM DMA moves the whole 85x128 tile into LDS ----
    // TDM issues once per wave and ignores EXEC -> only wave 0 issues it.
    if (tid < 32u) {
        // low 32 bits of a generic LDS pointer == byte offset within wave LDS
        const uint32_t lds_base = (uint32_t)(uintptr_t)(&tile[0]);
        const uint64_t gaddr    = (uint64_t)(uintptr_t)in_plane;

        // D# group 0: count=1 | lds_addr | global_addr[56:0] | type=2
        const v4u g0 = {
            1u,                                                  // count=1, no gather
            lds_base,                                            // lds_addr (bytes)
            (uint32_t)gaddr,                                     // global_addr[31:0]
            (((uint32_t)(gaddr >> 32)) & 0x01FFFFFFu) | (2u << 30) // addr[56:32] | type=2
        };

        // D# group 1:
        //  w0: wg_mask=0 | data_size=2 (4B) | pad_enable=1 | pad_interval=6 (128 dw)
        //      | pad_amount=0 (1 dw)  => LDS row stride = 129 dwords
        //  tensor_dim0=6400, tensor_dim1=85, tile_dim0=128, tile_dim1=85,
        //  tensor_dim0_stride=6400 (elements)
        const v8u g1 = {
            (2u << 16) | (1u << 20) | (6u << 22),   // data_size/pad config
            ((uint32_t)HW & 0xFFFFu) << 16,         // tensor_dim0[15:0] @ bits 63:48
            ((uint32_t)ATTRS) << 16,                // tensor_dim1[15:0] @ bits 95:80
            ((uint32_t)TP) << 16,                   // tile_dim0        @ bits 127:112
            (uint32_t)ATTRS,                        // tile_dim1        @ bits 143:128
            (uint32_t)HW,                           // tensor_dim0_stride[31:0]
            0u,                                     // stride hi / dim1_stride lo
            0u                                      // dim1_stride hi
        };

        // 2D tensor: VADDR2/VADDR3 omitted (NULL)
        asm volatile("tensor_load_to_lds %0, %1"
                     :: "s"(g0), "s"(g1)
                     : "memory");
        __builtin_amdgcn_s_wait_tensorcnt(0);
    }
    __syncthreads();

    // anchor dims (uniform scalar loads)
    const float aw = anchors[a * 2 + 0];
    const float ah = anchors[a * 2 + 1];

    // output tile base: flat index = (b*19200 + a*6400 + pixel0)*85
    float* out_base =
        out + ((size_t)(b * (NANCH * HW) + a * HW + pixel0)) * ATTRS;

    // ---- Phase 2: transposed LDS read, per-attr transform, coalesced NT store ----
    // flat out idx = pix*85 + attr; idx += 256 => attr += 1, pix += 3 (256 = 3*85+1)
    uint32_t attr = tid % ATTRS;
    uint32_t pix  = tid / ATTRS;
    for (uint32_t idx = tid; idx < ATTRS * TP; idx += BLOCK) {
        float v = tile[attr * LDS_STRIDE + pix];
        float r;
        if (attr >= 4u) {
            // conf + class scores
            r = fast_sigmoid(v);
        } else if (attr < 2u) {
            // box center: (sigmoid + grid) * stride
            uint32_t gp = pixel0 + pix;
            float g = (attr == 0u) ? (float)(gp % GW) : (float)(gp / GW);
            r = (fast_sigmoid(v) + g) * STRIDE_PX;
        } else {
            // box size: exp(clip(v, -16, 16)) * anchor
            float cv = fminf(fmaxf(v, -16.0f), 16.0f);
            r = __expf(cv) * ((attr == 2u) ? aw : ah);
        }
        __builtin_nontemporal_store(r, out_base + idx);

        attr += 1u; pix += 3u;
        if (attr >= ATTRS) { attr -= ATTRS; pix += 1u; }
    }
}

extern "C" void kernel_launch(void* const* d_in, const int* in_sizes, int n_in,
                              void* d_out, int out_size, void* d_ws, size_t ws_size,
                              hipStream_t stream) {
    const float* in      = (const float*)d_in[0];   // [32,255,80,80] f32
    const float* anchors = (const float*)d_in[1];   // [3,2] f32
    float* out           = (float*)d_out;           // [32,19200,85] f32

    dim3 grid(BATCH * NANCH * TILES_PP);            // 4800 blocks
    yolo_head_kernel<<<grid, BLOCK, 0, stream>>>(in, anchors, out);
    (void)in_sizes; (void)n_in; (void)out_size; (void)d_ws; (void)ws_size;
}